// YOLOv1LossFunction_36464272343376
// MI455X (gfx1250) — compile-verified
//
#include <hip/hip_runtime.h>
#include <stdint.h>

// ---------------------------------------------------------------------------
// YOLOv1 loss on MI455X (gfx1250).
// Bandwidth-bound (160 MB traffic, ~0.12 GFLOP): stage slabs into LDS with
// CDNA5 async global->LDS B128 DMA, compute per-cell loss from LDS, reduce
// per-wave with v_wmma_f32_16x16x4_f32 (B = ones => exact sum), then a
// deterministic single-block final reduction (no float atomics).
// ---------------------------------------------------------------------------

typedef __attribute__((ext_vector_type(2))) float v2f;
typedef __attribute__((ext_vector_type(8))) float v8f;

#define THREADS          128
#define CELLS_PER_BLOCK  128
#define CELL_F           50                                  // floats per cell per tensor
#define STAGE_F4         (CELLS_PER_BLOCK * CELL_F / 4)      // 1600 float4 per tensor
#define SLAB_F           (CELLS_PER_BLOCK * CELL_F)          // 6400 floats

__device__ __forceinline__ float iou_f(const float a[4], const float b[4]) {
    float w = fminf(a[0] + a[2] * 0.5f, b[0] + b[2] * 0.5f)
            - fmaxf(a[0] - a[2] * 0.5f, b[0] - b[2] * 0.5f);
    float h = fminf(a[1] + a[3] * 0.5f, b[1] + b[3] * 0.5f)
            - fmaxf(a[1] - a[3] * 0.5f, b[1] - b[3] * 0.5f);
    float cross = w * h;
    float iou = cross / (a[2] * a[3] + b[2] * b[3] - cross);
    return (w <= 0.0f || h <= 0.0f) ? 0.0f : iou;
}

__global__ __launch_bounds__(THREADS)
void yolo_cell_kernel(const float* __restrict__ pr, const float* __restrict__ gt,
                      float* __restrict__ ws, long total_cells) {
    __shared__ __align__(16) float sP[SLAB_F];
    __shared__ __align__(16) float sG[SLAB_F];
    __shared__ float red[THREADS / 32];

    const long base_cell = (long)blockIdx.x * CELLS_PER_BLOCK;

    // ---- async DMA global -> LDS (CDNA5 path, tracked by ASYNCcnt) ----
    {
        const uint64_t pr_base = (uint64_t)(uintptr_t)pr;
        const uint64_t gt_base = (uint64_t)(uintptr_t)gt;
        const uint64_t slab_off = (uint64_t)base_cell * (CELL_F * 4);
        const uint64_t max_off  = (uint64_t)total_cells * (CELL_F * 4) - 16;  // clamp for tail
        uint32_t ldsP = (uint32_t)(uintptr_t)&sP[0];
        uint32_t ldsG = (uint32_t)(uintptr_t)&sG[0];
        for (int i = threadIdx.x; i < STAGE_F4; i += THREADS) {
            uint64_t off = slab_off + (uint64_t)i * 16;
            if (off > max_off) off = max_off;                 // duplicate tail reads, unused
            uint64_t ap = pr_base + off;
            uint64_t ag = gt_base + off;
            uint32_t lp = ldsP + (uint32_t)i * 16;
            uint32_t lg = ldsG + (uint32_t)i * 16;
            asm volatile("global_load_async_to_lds_b128 %0, %1, off"
                         :: "v"(lp), "v"(ap) : "memory");
            asm volatile("global_load_async_to_lds_b128 %0, %1, off"
                         :: "v"(lg), "v"(ag) : "memory");
        }
        asm volatile("s_wait_asynccnt 0x0" ::: "memory");
    }
    __syncthreads();

    // ---- per-cell loss (exactly mirrors the reference semantics) ----
    const long  c     = base_cell + (long)threadIdx.x;
    const bool  valid = (c < total_cells);
    const float* P = &sP[threadIdx.x * CELL_F];
    const float* G = &sG[threadIdx.x * CELL_F];

    float loss = 0.0f;
    const float gc0 = G[0];
    const float gc1 = G[25];
    const int obj = (gc0 != 0.0f) + (gc1 != 0.0f);
    const float pc0 = P[0];
    const float pc1 = P[25];

    if (obj == 0) {
        loss = 0.5f * (pc0 + pc1);                        // LAMBDA_NOOBJ * sum(pr_conf)
    } else {
        loss = (pc0 - 1.0f) * (pc0 - 1.0f) + (pc1 - 1.0f) * (pc1 - 1.0f);
        float gb0[4], gb1[4];
#pragma unroll
        for (int k = 0; k < 4; ++k) { gb0[k] = G[21 + k]; gb1[k] = G[46 + k]; }
#pragma unroll
        for (int i = 0; i < 2; ++i) {
            const float* Pi = P + i * 25;
            float pb[4];
#pragma unroll
            for (int k = 0; k < 4; ++k) pb[k] = Pi[21 + k];
            // valid = gt_idx < obj_num; first-max argmax (strict > to switch)
            int rsp = 0;
            const float i0 = iou_f(pb, gb0);
            if (obj == 2) {
                const float i1 = iou_f(pb, gb1);
                if (i1 > i0) rsp = 1;
            }
            float rb[4];
#pragma unroll
            for (int k = 0; k < 4; ++k) rb[k] = rsp ? gb1[k] : gb0[k];
            const float dx = pb[0] - rb[0];
            const float dy = pb[1] - rb[1];
            const float sw = sqrtf(pb[2]) - sqrtf(rb[2]);
            const float sh = sqrtf(pb[3]) - sqrtf(rb[3]);
            loss += 5.0f * (dx * dx + dy * dy + sw * sw + sh * sh);   // LAMBDA_COORD
            const float* rc = G + rsp * 25 + 1;                        // responsible gt cls
            float cl = 0.0f;
#pragma unroll
            for (int k = 0; k < 20; ++k) {
                const float d = Pi[1 + k] - rc[k];
                cl += d * d;
            }
            loss += cl * 0.05f;                                        // mean over 20
        }
    }
    if (!valid) loss = 0.0f;

    // ---- wave reduction via WMMA: A = partials, B = ones  =>  sum(D) = 16*sum(A).
    // Layout-agnostic (we sum every element of D), exact in f32 accumulation.
    // EXEC is all-ones here: all 128 lanes reconverged above. ----
    v2f a;  a[0] = loss; a[1] = 0.0f;
    v2f bo; bo[0] = 1.0f; bo[1] = 1.0f;
    v8f cz = {};
    v8f dd = __builtin_amdgcn_wmma_f32_16x16x4_f32(false, a, false, bo,
                                                   (short)0, cz, false, false);
    float s = dd[0] + dd[1] + dd[2] + dd[3] + dd[4] + dd[5] + dd[6] + dd[7];
#pragma unroll
    for (int off = 16; off > 0; off >>= 1) s += __shfl_xor(s, off, 32);
    const float wave_total = s * 0.0625f;   // /16

    if ((threadIdx.x & 31) == 0) red[threadIdx.x >> 5] = wave_total;
    __syncthreads();
    if (threadIdx.x == 0)
        ws[blockIdx.x] = red[0] + red[1] + red[2] + red[3];
}

__global__ __launch_bounds__(256)
void yolo_reduce_kernel(const float* __restrict__ ws, float* __restrict__ out,
                        int nblocks, float inv_batch) {
    __shared__ float sm[256];
    float s = 0.0f;
    for (int i = threadIdx.x; i < nblocks; i += 256) s += ws[i];   // fixed order: deterministic
    sm[threadIdx.x] = s;
    __syncthreads();
#pragma unroll
    for (int k = 128; k > 0; k >>= 1) {
        if ((int)threadIdx.x < k) sm[threadIdx.x] += sm[threadIdx.x + k];
        __syncthreads();
    }
    if (threadIdx.x == 0) out[0] = sm[0] * inv_batch;
}

extern "C" void kernel_launch(void* const* d_in, const int* in_sizes, int n_in,
                              void* d_out, int out_size, void* d_ws, size_t ws_size,
                              hipStream_t stream) {
    const float* pr = (const float*)d_in[0];
    const float* gt = (const float*)d_in[1];
    float* out = (float*)d_out;
    float* ws  = (float*)d_ws;

    const int D = 7 * 7 * 2 * 25;                  // 2450 floats per batch row
    const long batch = (long)(in_sizes[0] / D);    // 8192
    const long total_cells = batch * 49;           // 401408
    const int nblocks = (int)((total_cells + CELLS_PER_BLOCK - 1) / CELLS_PER_BLOCK);

    yolo_cell_kernel<<<nblocks, THREADS, 0, stream>>>(pr, gt, ws, total_cells);
    yolo_reduce_kernel<<<1, 256, 0, stream>>>(ws, out, nblocks, 1.0f / (float)batch);
}